// FinalTrans_61426622267536
// MI455X (gfx1250) — compile-verified
//
#include <hip/hip_runtime.h>
#include <math.h>

#define BATCH 16
#define NPTS  2048
#define STRIPS_PER_BATCH (NPTS / 16)            // 128
#define TOTAL_STRIPS (BATCH * STRIPS_PER_BATCH) // 2048
#define EPSF 1e-6f
#define SIGMA_INV2 (1.0f / 100.0f)  // 1/SIGMA_D^2
#define INLIER_THR_F 4.0f
#define POWER_ITERS 10
#define REFINE_STEPS 5

typedef float v2f __attribute__((ext_vector_type(2)));
typedef float v8f __attribute__((ext_vector_type(8)));

#if defined(__has_builtin)
#  if __has_builtin(__builtin_amdgcn_wmma_f32_16x16x4_f32)
#    define HAVE_WMMA_F32X4 1
#  endif
#endif

// Raw v_sqrt_f32 (~1 ulp), skipping the IEEE software-refinement fixup that
// libm sqrtf emits. Distances here are O(1..1e3) vs SIGMA_D=10 tolerance.
__device__ __forceinline__ float fast_sqrtf(float x) {
#if defined(__AMDGCN__)
    return __builtin_amdgcn_sqrtf(x);
#else
    return sqrtf(x);
#endif
}

// ---------------------------------------------------------------------------
// init: v0 = ones(B*N)
// ---------------------------------------------------------------------------
__global__ void pw_init_kernel(float* v0) {
    int i = blockIdx.x * blockDim.x + threadIdx.x;
    if (i < BATCH * NPTS) v0[i] = 1.0f;
}

// ---------------------------------------------------------------------------
// One power iteration: y = M * (v_in * s),  s = 1/(sqrt(prev ||y||^2)+eps).
// M is the compat matrix recomputed on the fly; 16x16 Gram tiles via
// V_WMMA_F32_16X16X4_F32 (K=2 coords zero-padded to 4).
// One wave (32 lanes) per 16-row strip. Deterministic reductions only.
// Diagonal handled by subtracting v_i*s at the write stage (c_ii computes ~1).
// ---------------------------------------------------------------------------
__global__ void pw_matvec_kernel(const float2* __restrict__ src,
                                 const float2* __restrict__ tgt,
                                 const float* __restrict__ v_in,
                                 float* __restrict__ v_out,
                                 const float* __restrict__ p_in, // per-strip Sum y^2 (prev iter)
                                 float* __restrict__ p_out,
                                 int first) {
    const int strip = blockIdx.x * (blockDim.x >> 5) + (threadIdx.x >> 5);
    const int lane  = threadIdx.x & 31;
    const int b     = strip >> 7;              // strip / 128
    const int srow  = (strip & 127) << 4;      // 16-row strip base

    const float2* As = src + b * NPTS;
    const float2* At = tgt + b * NPTS;
    const float*  vb = v_in + b * NPTS;

    // lazy normalization scale from previous iteration's partials (deterministic order)
    float s;
    if (first) {
        s = 1.0f;
    } else {
        float nrm = 0.0f;
        const float* pb = p_in + (b << 7);
        for (int i = 0; i < STRIPS_PER_BATCH; ++i) nrm += pb[i];
        s = 1.0f / (sqrtf(nrm) + EPSF);
    }

    const bool lo  = lane < 16;
    const int  hi8 = lo ? 0 : 8;

    // A operand: lane m (0..15) holds row srow+m coords in K=0,1; lanes 16..31 -> K=2,3 = 0
    float2 rowS = As[srow + (lane & 15)];
    float2 rowT = At[srow + (lane & 15)];
    v2f a_s, a_t;
    a_s.x = lo ? rowS.x : 0.0f;  a_s.y = lo ? rowS.y : 0.0f;
    a_t.x = lo ? rowT.x : 0.0f;  a_t.y = lo ? rowT.y : 0.0f;

    // squared norms of this strip's 16 rows, broadcast to the entry layout of D
    float rn_s = rowS.x * rowS.x + rowS.y * rowS.y;
    float rn_t = rowT.x * rowT.x + rowT.y * rowT.y;
    float nis[8], nit[8];
#pragma unroll
    for (int e = 0; e < 8; ++e) {
        nis[e] = __shfl(rn_s, e + hi8, 32);
        nit[e] = __shfl(rn_t, e + hi8, 32);
    }

    float acc[8];
#pragma unroll
    for (int e = 0; e < 8; ++e) acc[e] = 0.0f;

    for (int j0 = 0; j0 < NPTS; j0 += 16) {
        const int jl = j0 + (lane & 15);
        float2 ps = As[jl];
        float2 pt = At[jl];
        float  vj = vb[jl] * s;
        float nsj = ps.x * ps.x + ps.y * ps.y;
        float ntj = pt.x * pt.x + pt.y * pt.y;

        // B operand: K=0,1 rows in lanes 0..15; K=2,3 rows (lanes 16..31) = 0
        v2f b_s, b_t;
        b_s.x = lo ? ps.x : 0.0f;  b_s.y = lo ? ps.y : 0.0f;
        b_t.x = lo ? pt.x : 0.0f;  b_t.y = lo ? pt.y : 0.0f;

        v8f gs, gt;
#if defined(HAVE_WMMA_F32X4)
        v8f c0 = {};
        gs = __builtin_amdgcn_wmma_f32_16x16x4_f32(false, a_s, false, b_s,
                                                   (short)0, c0, false, false);
        gt = __builtin_amdgcn_wmma_f32_16x16x4_f32(false, a_t, false, b_t,
                                                   (short)0, c0, false, false);
#else
        // VALU fallback: gram entries via shuffled row coords
#pragma unroll
        for (int e = 0; e < 8; ++e) {
            float rxs = __shfl(rowS.x, e + hi8, 32), rys = __shfl(rowS.y, e + hi8, 32);
            float rxt = __shfl(rowT.x, e + hi8, 32), ryt = __shfl(rowT.y, e + hi8, 32);
            gs[e] = fmaf(rxs, ps.x, rys * ps.y);
            gt[e] = fmaf(rxt, pt.x, ryt * pt.y);
        }
#endif

#pragma unroll
        for (int e = 0; e < 8; ++e) {
            float d2s  = fmaxf(nis[e] + nsj - 2.0f * gs[e], 0.0f);
            float d2t  = fmaxf(nit[e] + ntj - 2.0f * gt[e], 0.0f);
            float diff = fast_sqrtf(d2s) - fast_sqrtf(d2t);
            float c    = fmaxf(fmaf(diff * diff, -SIGMA_INV2, 1.0f), 0.0f);
            acc[e]     = fmaf(c, vj, acc[e]);   // diagonal corrected after reduce
        }
    }

    // reduce each D entry across its 16 columns (xor network stays within halves)
#pragma unroll
    for (int e = 0; e < 8; ++e) {
        float a = acc[e];
        a += __shfl_xor(a, 1, 32);
        a += __shfl_xor(a, 2, 32);
        a += __shfl_xor(a, 4, 32);
        a += __shfl_xor(a, 8, 32);
        acc[e] = a;
    }

    float sq = 0.0f;
    if (lane == 0 || lane == 16) {
        float* yo = v_out + b * NPTS + srow + hi8;
        const float* vd = vb + srow + hi8;
#pragma unroll
        for (int e = 0; e < 8; ++e) {
            float yv = acc[e] - vd[e] * s;      // remove diagonal term (c_ii ~= 1)
            yo[e] = yv;
            sq = fmaf(yv, yv, sq);
        }
    }
    sq += __shfl_xor(sq, 16, 32);              // lane0 gets full strip sum
    if (lane == 0) p_out[strip] = sq;
}

// ---------------------------------------------------------------------------
// block reduction, deterministic, broadcasts result to all threads (256 thr)
// ---------------------------------------------------------------------------
__device__ __forceinline__ float block_reduce(float v, float* sbuf) {
#pragma unroll
    for (int m = 16; m >= 1; m >>= 1) v += __shfl_xor(v, m, 32);
    __syncthreads();                           // protect sbuf reuse
    if ((threadIdx.x & 31) == 0) sbuf[threadIdx.x >> 5] = v;
    __syncthreads();
    float r = sbuf[0];
#pragma unroll
    for (int i = 1; i < 8; ++i) r += sbuf[i];  // same order on every thread
    return r;
}

__device__ __forceinline__ float refine_weight(float2 a, float2 bb,
                                               float r00, float r01, float r10, float r11,
                                               float tx, float ty) {
    float px = fmaf(r00, a.x, fmaf(r01, a.y, tx));
    float py = fmaf(r10, a.x, fmaf(r11, a.y, ty));
    float dx = px - bb.x, dy = py - bb.y;
    float L2 = sqrtf(fmaf(dx, dx, dy * dy));
    if (L2 >= INLIER_THR_F) return 0.0f;
    float q = L2 * (1.0f / INLIER_THR_F);
    return 1.0f / (1.0f + q * q);
}

// ---------------------------------------------------------------------------
// Per-batch weighted Procrustes (closed-form 2D rotation) + 5 IRLS steps,
// then write transformed src. One 256-thread block per batch.
// ---------------------------------------------------------------------------
__global__ void procrustes_kernel(const float2* __restrict__ src,
                                  const float2* __restrict__ tgt,
                                  const float* __restrict__ y,   // un-normalized eigvec
                                  float2* __restrict__ out) {
    __shared__ float sbuf[8];
    const int b = blockIdx.x;
    const int t = threadIdx.x;
    const float2* A  = src + b * NPTS;
    const float2* Bp = tgt + b * NPTS;
    const float*  yb = y + b * NPTS;

    // w = v/(sum(v)+eps), v = y/(||y||+eps)
    float sq = 0.0f, sm = 0.0f;
    for (int n = t; n < NPTS; n += 256) {
        float v = yb[n];
        sq = fmaf(v, v, sq);
        sm += v;
    }
    sq = block_reduce(sq, sbuf);
    sm = block_reduce(sm, sbuf);
    const float s     = 1.0f / (sqrtf(sq) + EPSF);
    const float denom = 1.0f / (sm * s + EPSF);

    float r00 = 1.0f, r01 = 0.0f, r10 = 0.0f, r11 = 1.0f, tx = 0.0f, ty = 0.0f;

    for (int step = 0; step <= REFINE_STEPS; ++step) {
        // pass 1: weighted centroids
        float wsum = 0.0f, wax = 0.0f, way = 0.0f, wbx = 0.0f, wby = 0.0f;
        for (int n = t; n < NPTS; n += 256) {
            float2 a = A[n], bb = Bp[n];
            float w = (step == 0) ? (yb[n] * s * denom)
                                  : refine_weight(a, bb, r00, r01, r10, r11, tx, ty);
            wsum += w;
            wax = fmaf(w, a.x, wax);  way = fmaf(w, a.y, way);
            wbx = fmaf(w, bb.x, wbx); wby = fmaf(w, bb.y, wby);
        }
        wsum = block_reduce(wsum, sbuf);
        wax = block_reduce(wax, sbuf);  way = block_reduce(way, sbuf);
        wbx = block_reduce(wbx, sbuf);  wby = block_reduce(wby, sbuf);
        float inv = 1.0f / (wsum + EPSF);
        float cax = wax * inv, cay = way * inv, cbx = wbx * inv, cby = wby * inv;

        // pass 2: covariance H on centered coords
        float h00 = 0.0f, h01 = 0.0f, h10 = 0.0f, h11 = 0.0f;
        for (int n = t; n < NPTS; n += 256) {
            float2 a = A[n], bb = Bp[n];
            float w = (step == 0) ? (yb[n] * s * denom)
                                  : refine_weight(a, bb, r00, r01, r10, r11, tx, ty);
            float amx = a.x - cax, amy = a.y - cay;
            float bmx = bb.x - cbx, bmy = bb.y - cby;
            h00 = fmaf(w * amx, bmx, h00);
            h01 = fmaf(w * amx, bmy, h01);
            h10 = fmaf(w * amy, bmx, h10);
            h11 = fmaf(w * amy, bmy, h11);
        }
        h00 = block_reduce(h00, sbuf); h01 = block_reduce(h01, sbuf);
        h10 = block_reduce(h10, sbuf); h11 = block_reduce(h11, sbuf);

        // closed-form best proper 2D rotation (== SVD + det fix)
        float theta = atan2f(h01 - h10, h00 + h11);
        float c = cosf(theta), sn = sinf(theta);
        r00 = c; r01 = -sn; r10 = sn; r11 = c;
        tx = cbx - (r00 * cax + r01 * cay);
        ty = cby - (r10 * cax + r11 * cay);
    }

    float2* ob = out + b * NPTS;
    for (int n = t; n < NPTS; n += 256) {
        float2 a = A[n];
        float2 p;
        p.x = fmaf(r00, a.x, fmaf(r01, a.y, tx));
        p.y = fmaf(r10, a.x, fmaf(r11, a.y, ty));
        ob[n] = p;
    }
}

// ---------------------------------------------------------------------------
extern "C" void kernel_launch(void* const* d_in, const int* in_sizes, int n_in,
                              void* d_out, int out_size, void* d_ws, size_t ws_size,
                              hipStream_t stream) {
    (void)in_sizes; (void)n_in; (void)out_size; (void)ws_size;
    const float2* src = (const float2*)d_in[0];
    const float2* tgt = (const float2*)d_in[1];

    float* ws = (float*)d_ws;
    const int BN = BATCH * NPTS;               // 32768
    float* v0 = ws;                            // ping
    float* v1 = ws + BN;                       // pong
    float* p0 = ws + 2 * BN;                   // per-strip ||y||^2 partials ping
    float* p1 = p0 + TOTAL_STRIPS;             // pong   (total ws: ~278 KB)

    pw_init_kernel<<<(BN + 255) / 256, 256, 0, stream>>>(v0);

    for (int k = 0; k < POWER_ITERS; ++k) {
        float* vin  = (k & 1) ? v1 : v0;
        float* vout = (k & 1) ? v0 : v1;
        float* pin  = (k & 1) ? p1 : p0;
        float* pout = (k & 1) ? p0 : p1;
        pw_matvec_kernel<<<TOTAL_STRIPS / 4, 128, 0, stream>>>(
            src, tgt, vin, vout, pin, pout, (k == 0) ? 1 : 0);
    }
    // after 10 iters (even), final y is in v0
    procrustes_kernel<<<BATCH, 256, 0, stream>>>(src, tgt, v0, (float2*)d_out);
}